// VectorQuantizer_66305705115810
// MI455X (gfx1250) — compile-verified
//
#include <hip/hip_runtime.h>
#include <hip/hip_bf16.h>

typedef __attribute__((ext_vector_type(2))) float    v2f;
typedef __attribute__((ext_vector_type(8))) float    v8f;
typedef __attribute__((ext_vector_type(4))) unsigned v4u;
typedef __attribute__((ext_vector_type(4))) int      v4i;
typedef __attribute__((ext_vector_type(8))) int      v8i;

#define NUM_EMB 512
#define EMB_DIM 64
#define N_ROWS  65536       // 64*32*32
#define LDS_STRIDE 68       // 64 + 4 pad floats -> conflict-free ds_load_b64

// ---------------------------------------------------------------------------
// Prep: half_norm[k] = 0.5*||e_k||^2, cbT[k][d] = codebook[d][k],
//       zero histogram + sq-error accumulator (determinism per launch).
// ---------------------------------------------------------------------------
__global__ void vq_prep_kernel(const float* __restrict__ cb,
                               float* __restrict__ half_norm,
                               float* __restrict__ cbT,
                               unsigned* __restrict__ counts,
                               float* __restrict__ sqacc) {
    int k = blockIdx.x * blockDim.x + threadIdx.x;
    if (k < NUM_EMB) {
        float s = 0.f;
#pragma unroll
        for (int d = 0; d < EMB_DIM; ++d) {
            float v = cb[d * NUM_EMB + k];
            s += v * v;
            cbT[k * EMB_DIM + d] = v;
        }
        half_norm[k] = 0.5f * s;
        counts[k] = 0u;
    }
    if (k == 0) *sqacc = 0.f;
}

// ---------------------------------------------------------------------------
// Main: scores = x.E - 0.5||e||^2 via V_WMMA_F32_16X16X4_F32 with the
// codebook staged into LDS by the Tensor Data Mover (padded rows, bank-
// conflict-free). Block = 256 threads (8 waves); each wave owns two 16-row
// tiles so every B fragment is reused twice.
// ---------------------------------------------------------------------------
__global__ void __launch_bounds__(256)
vq_main_kernel(const float* __restrict__ x,
               const float* __restrict__ half_norm,
               const float* __restrict__ cbT,
               float* __restrict__ out,
               unsigned* __restrict__ counts,
               float* __restrict__ sqacc) {
    __shared__ int   idx_lds[256];
    __shared__ float red_lds[8];
    extern __shared__ float lds_dyn[];                 // [512*68] cb + [512] hn
    float* lds_cb = lds_dyn;
    float* lds_hn = lds_dyn + NUM_EMB * LDS_STRIDE;

    const int lane = threadIdx.x & 31;
    const int wave = threadIdx.x >> 5;
    const int h    = lane >> 4;   // half-wave select (K split for A/B frags)
    const int m    = lane & 15;

    // ---- stage half_norm (all threads) -----------------------------------
    for (int i = threadIdx.x; i < NUM_EMB; i += 256) lds_hn[i] = half_norm[i];

    // ---- stage codebook: one TDM descriptor DMAs 512x64 f32 global ->
    //      LDS with 4-dword pad after every 64 dwords (row stride 68).
#if defined(__has_builtin) && __has_builtin(__builtin_amdgcn_tensor_load_to_lds)
    if (threadIdx.x < 32) {
        const unsigned           ldsoff = (unsigned)(size_t)(void*)lds_cb;
        const unsigned long long ga     = (unsigned long long)(size_t)cbT;
        v4u g0;
        g0.x = 1u;                                        // count=1, user D#
        g0.y = ldsoff;                                    // lds_addr
        g0.z = (unsigned)(ga & 0xffffffffu);              // global_addr lo
        g0.w = (unsigned)((ga >> 32) & 0x01ffffffu)       // global_addr hi
             | (2u << 30);                                // type = image
        v8i g1;
        g1[0] = (int)((2u << 16)        // data_size = 4B
                    | (1u << 20)        // pad_enable
                    | (5u << 22)        // pad_interval: 64 dwords
                    | (3u << 25));      // pad_amount:   4 dwords
        g1[1] = (int)(64u  << 16);      // tensor_dim0 = 64
        g1[2] = (int)(512u << 16);      // tensor_dim1 = 512
        g1[3] = (int)(64u  << 16);      // tile_dim0   = 64
        g1[4] = (int)512;               // tile_dim1   = 512 (tile_dim2 = 0)
        g1[5] = 64;                     // tensor_dim0_stride = 64
        g1[6] = 0;
        g1[7] = 0;
        v4i gz = {0, 0, 0, 0};
#if __clang_major__ >= 23
        v8i gz8 = {0, 0, 0, 0, 0, 0, 0, 0};
        __builtin_amdgcn_tensor_load_to_lds(g0, g1, gz, gz, gz8, 0);
#else
        __builtin_amdgcn_tensor_load_to_lds(g0, g1, gz, gz, 0);
#endif
        __builtin_amdgcn_s_wait_tensorcnt(0);
    }
#else
    // fallback: staged copy with the same padded layout
    for (int i4 = threadIdx.x; i4 < NUM_EMB * EMB_DIM / 4; i4 += 256) {
        const int g = i4 * 4;
        const int k = g >> 6, d = g & 63;
        float4 v = *(const float4*)(cbT + g);
        *(float4*)(lds_cb + k * LDS_STRIDE + d) = v;
    }
#endif
    __syncthreads();

    const int block_row0 = blockIdx.x * 256;
    const int row0 = block_row0 + wave * 32;  // tile0: rows row0..row0+15
    const int row1 = row0 + 16;               // tile1

    // ---- preload A fragments: A is 16x4 f32 per step; lane m = row,
    //      lanes 0-15 hold K=4s+{0,1}, lanes 16-31 hold K=4s+{2,3}.
    v2f a0[16], a1[16];
    {
        const float* r0 = x + (size_t)(row0 + m) * EMB_DIM;
        const float* r1 = x + (size_t)(row1 + m) * EMB_DIM;
#pragma unroll
        for (int s = 0; s < 16; ++s) {
            const int off = 4 * s + 2 * h;
            a0[s] = *(const v2f*)(r0 + off);
            a1[s] = *(const v2f*)(r1 + off);
        }
    }

    float bv0[8], bv1[8];
    int   bi0[8], bi1[8];
#pragma unroll
    for (int j = 0; j < 8; ++j) {
        bv0[j] = -3.4e38f; bv1[j] = -3.4e38f; bi0[j] = 0; bi1[j] = 0;
    }

    // ---- sweep 32 column tiles of 16 codes each --------------------------
    for (int t = 0; t < NUM_EMB / 16; ++t) {
        const int col = t * 16 + m;            // C layout: lane%16 = column
        const float hn = lds_hn[col];

        // batch all 16 B fragments from LDS (conflict-free ds_load_b64)
        const float* lb = lds_cb + col * LDS_STRIDE + 2 * h;
        v2f bf[16];
#pragma unroll
        for (int s = 0; s < 16; ++s) bf[s] = *(const v2f*)(lb + 4 * s);

        v8f c0, c1;
#pragma unroll
        for (int j = 0; j < 8; ++j) { c0[j] = -hn; c1[j] = -hn; }

#pragma unroll
        for (int s = 0; s < 16; ++s) {
            c0 = __builtin_amdgcn_wmma_f32_16x16x4_f32(
                     false, a0[s], false, bf[s], (short)0, c0, false, false);
            c1 = __builtin_amdgcn_wmma_f32_16x16x4_f32(
                     false, a1[s], false, bf[s], (short)0, c1, false, false);
        }
        // running argmax (strict > keeps earliest column on ties)
#pragma unroll
        for (int j = 0; j < 8; ++j) {
            if (c0[j] > bv0[j]) { bv0[j] = c0[j]; bi0[j] = col; }
            if (c1[j] > bv1[j]) { bv1[j] = c1[j]; bi1[j] = col; }
        }
    }

    // ---- butterfly argmax across the 16 lanes sharing each row -----------
#pragma unroll
    for (int off = 8; off >= 1; off >>= 1) {
#pragma unroll
        for (int j = 0; j < 8; ++j) {
            float ov = __shfl_xor(bv0[j], off, 16);
            int   oi = __shfl_xor(bi0[j], off, 16);
            if (ov > bv0[j] || (ov == bv0[j] && oi < bi0[j])) { bv0[j] = ov; bi0[j] = oi; }
            ov = __shfl_xor(bv1[j], off, 16);
            oi = __shfl_xor(bi1[j], off, 16);
            if (ov > bv1[j] || (ov == bv1[j] && oi < bi1[j])) { bv1[j] = ov; bi1[j] = oi; }
        }
    }

    // C row mapping: row_in_tile = j + 8*h -> lane 0 / lane 16 publish
    if (m == 0) {
#pragma unroll
        for (int j = 0; j < 8; ++j) {
            idx_lds[wave * 32 +      8 * h + j] = bi0[j];
            idx_lds[wave * 32 + 16 + 8 * h + j] = bi1[j];
        }
    }
    __syncthreads();

    // ---- output phase: one thread per row --------------------------------
    const int   row = block_row0 + threadIdx.x;
    const int   idx = idx_lds[threadIdx.x];
    const float* q   = cbT + (size_t)idx * EMB_DIM;
    const float* xin = x   + (size_t)row * EMB_DIM;
    float*       orw = out + (size_t)row * EMB_DIM;
    float lsum = 0.f;
#pragma unroll
    for (int d = 0; d < EMB_DIM; d += 4) {
        float4 qv = *(const float4*)(q + d);
        float4 xv = *(const float4*)(xin + d);
        *(float4*)(orw + d) = qv;   // ste == quantized numerically
        float e0 = qv.x - xv.x, e1 = qv.y - xv.y;
        float e2 = qv.z - xv.z, e3 = qv.w - xv.w;
        lsum += e0 * e0 + e1 * e1 + e2 * e2 + e3 * e3;
    }
    atomicAdd(&counts[idx], 1u);

    // block reduce squared error -> one global atomic per block
#pragma unroll
    for (int off = 16; off >= 1; off >>= 1) lsum += __shfl_xor(lsum, off, 32);
    if (lane == 0) red_lds[wave] = lsum;
    __syncthreads();
    if (threadIdx.x == 0) {
        float s = 0.f;
#pragma unroll
        for (int i = 0; i < 8; ++i) s += red_lds[i];
        atomicAdd(sqacc, s);
    }
}

// ---------------------------------------------------------------------------
// Epilogue: perplexity + losses into the 3 tail outputs.
// ---------------------------------------------------------------------------
__global__ void vq_final_kernel(const unsigned* __restrict__ counts,
                                const float* __restrict__ sqacc,
                                float* __restrict__ out_tail) {
    __shared__ float red_lds[16];
    const int t = threadIdx.x;          // 512 threads
    const int lane = t & 31, wave = t >> 5;
    float p = (float)counts[t] * (1.0f / (float)N_ROWS);
    float term = p * __logf(p + 1e-10f);
#pragma unroll
    for (int off = 16; off >= 1; off >>= 1) term += __shfl_xor(term, off, 32);
    if (lane == 0) red_lds[wave] = term;
    __syncthreads();
    if (t == 0) {
        float s = 0.f;
#pragma unroll
        for (int i = 0; i < 16; ++i) s += red_lds[i];
        const float mse = *sqacc * (1.0f / ((float)N_ROWS * (float)EMB_DIM));
        out_tail[0] = __expf(-s);       // perplexity
        out_tail[1] = mse;              // codebook loss
        out_tail[2] = 0.25f * mse;      // commitment loss (BETA = 0.25)
    }
}

// ---------------------------------------------------------------------------
extern "C" void kernel_launch(void* const* d_in, const int* in_sizes, int n_in,
                              void* d_out, int out_size, void* d_ws, size_t ws_size,
                              hipStream_t stream) {
    const float* x  = (const float*)d_in[0];   // [65536, 64]
    const float* cb = (const float*)d_in[1];   // [64, 512]
    float* out = (float*)d_out;                // [65536*64 + 3]

    // workspace layout
    float*    half_norm = (float*)d_ws;                         // 512 f32
    float*    cbT       = half_norm + NUM_EMB;                  // 512*64 f32
    unsigned* counts    = (unsigned*)(cbT + NUM_EMB * EMB_DIM); // 512 u32
    float*    sqacc     = (float*)(counts + NUM_EMB);           // 1 f32

    const size_t dyn_lds = (size_t)(NUM_EMB * LDS_STRIDE + NUM_EMB) * sizeof(float);

    vq_prep_kernel<<<2, 256, 0, stream>>>(cb, half_norm, cbT, counts, sqacc);
    vq_main_kernel<<<N_ROWS / 256, 256, dyn_lds, stream>>>(x, half_norm, cbT, out,
                                                           counts, sqacc);
    vq_final_kernel<<<1, NUM_EMB, 0, stream>>>(counts, sqacc,
                                               out + (size_t)N_ROWS * EMB_DIM);
}